// WarpUV_33182917329193
// MI455X (gfx1250) — compile-verified
//
#include <hip/hip_runtime.h>
#include <hip/hip_bf16.h>
#include <math.h>

// ---------------------------------------------------------------------------
// WarpUV: closest-point-on-mesh UV warp + winding-number signed distance.
// Compute-bound FP32 geometry kernel; mesh staged to LDS via CDNA5 TDM
// (tensor_load_to_lds + s_wait_tensorcnt).
// ---------------------------------------------------------------------------

#define EPSF 1e-12f
#define MAX_NV 778
#define MAX_NF 1538
#define THREADS 256
#define PPT 2

typedef __attribute__((ext_vector_type(4))) unsigned int u32x4;
typedef __attribute__((ext_vector_type(8))) int          i32x8;
typedef __attribute__((ext_vector_type(4))) int          i32x4;

struct F3 { float x, y, z; };

__device__ __forceinline__ F3 f3(float x, float y, float z) { F3 r; r.x = x; r.y = y; r.z = z; return r; }
__device__ __forceinline__ F3 sub3(F3 a, F3 b) { return f3(a.x - b.x, a.y - b.y, a.z - b.z); }
__device__ __forceinline__ float dot3(F3 a, F3 b) { return a.x * b.x + a.y * b.y + a.z * b.z; }
__device__ __forceinline__ F3 mad3(F3 a, F3 d, float t) { return f3(a.x + t * d.x, a.y + t * d.y, a.z + t * d.z); }
__device__ __forceinline__ float safef(float d) { return fabsf(d) > EPSF ? d : 1.0f; }
__device__ __forceinline__ float clamp01(float v) { return fminf(fmaxf(v, 0.0f), 1.0f); }

// Ericson closest-point-on-triangle, with region priority matching the
// reference's jnp.where chain (last where wins => A > B > AB > C > AC > BC > interior).
__device__ __forceinline__ F3 closest_pt(F3 p, F3 a, F3 b, F3 c) {
    F3 ab = sub3(b, a), ac = sub3(c, a), ap = sub3(p, a);
    float d1 = dot3(ab, ap), d2 = dot3(ac, ap);
    F3 bp = sub3(p, b);
    float d3 = dot3(ab, bp), d4 = dot3(ac, bp);
    F3 cp = sub3(p, c);
    float d5 = dot3(ab, cp), d6 = dot3(ac, cp);
    float va = d3 * d6 - d5 * d4;
    float vb = d5 * d2 - d1 * d6;
    float vc = d1 * d4 - d3 * d2;
    if (d1 <= 0.0f && d2 <= 0.0f) return a;                                   // vertex A
    if (d3 >= 0.0f && d4 <= d3) return b;                                     // vertex B
    if (vc <= 0.0f && d1 >= 0.0f && d3 <= 0.0f)                               // edge AB
        return mad3(a, ab, d1 / safef(d1 - d3));
    if (d6 >= 0.0f && d5 <= d6) return c;                                     // vertex C
    if (vb <= 0.0f && d2 >= 0.0f && d6 <= 0.0f)                               // edge AC
        return mad3(a, ac, d2 / safef(d2 - d6));
    float e1 = d4 - d3, e2 = d5 - d6;
    if (va <= 0.0f && e1 >= 0.0f && e2 >= 0.0f)                               // edge BC
        return mad3(b, sub3(c, b), e1 / safef(e1 + e2));
    float den = safef(va + vb + vc);                                          // interior
    float v = vb / den, w = vc / den;
    return f3(a.x + v * ab.x + w * ac.x,
              a.y + v * ab.y + w * ac.y,
              a.z + v * ab.z + w * ac.z);
}

// One triangle's solid-angle contribution: atan2(num, den).
__device__ __forceinline__ float wind_term(F3 p, F3 a, F3 b, F3 c) {
    F3 ra = sub3(a, p), rb = sub3(b, p), rc = sub3(c, p);
    float la = sqrtf(dot3(ra, ra) + EPSF);
    float lb = sqrtf(dot3(rb, rb) + EPSF);
    float lc = sqrtf(dot3(rc, rc) + EPSF);
    F3 cr = f3(rb.y * rc.z - rb.z * rc.y,
               rb.z * rc.x - rb.x * rc.z,
               rb.x * rc.y - rb.y * rc.x);
    float num = dot3(ra, cr);
    float den = la * lb * lc + dot3(ra, rb) * lc + dot3(rb, rc) * la + dot3(rc, ra) * lb;
    return atan2f(num, den);
}

// Low 32 bits of a flat pointer into LDS == LDS byte address (shared aperture
// lives entirely in the high 32 bits; ISA 10.2).
__device__ __forceinline__ unsigned lds_addr_of(const void* p) {
    return (unsigned)(unsigned long long)(size_t)p;
}

#if __has_builtin(__builtin_amdgcn_tensor_load_to_lds) && __has_builtin(__builtin_amdgcn_s_wait_tensorcnt)
#define HAVE_TDM 1
#endif

#ifdef HAVE_TDM
// 1-D contiguous dword copy Global -> LDS via the Tensor Data Mover.
// D# per CDNA5 ISA 8.3-8.5: count=1, type=2, data_size=4B, tile_dim0=ndw,
// tensor_dim1=tile_dim1=1, workgroup_mask=0 (not in a cluster).
// All required state is in groups 0/1; groups 2/3 and the trailing vector
// (amdgpu-toolchain 6-arg form) are only meaningful for 3D-5D/gather tiles,
// so they are zero-filled.
__device__ __forceinline__ void tdm_load_dwords_1d(unsigned lds_addr, const void* gsrc, unsigned ndw) {
    unsigned long long ga = (unsigned long long)(size_t)gsrc;
    u32x4 g0;
    g0[0] = 1u;                                                    // count=1 (valid user D#)
    g0[1] = lds_addr;                                              // lds_addr [63:32]
    g0[2] = (unsigned)ga;                                          // global_addr[31:0]
    g0[3] = (unsigned)((ga >> 32) & 0x01FFFFFFull) | (2u << 30);   // global_addr[56:32] | type=2
    i32x8 g1;
    g1[0] = (int)(2u << 16);                                       // data_size=4B; wg_mask=0
    g1[1] = (int)((ndw & 0xFFFFu) << 16);                          // tensor_dim0[15:0] @63:48
    g1[2] = (int)(((ndw >> 16) & 0xFFFFu) | (1u << 16));           // tensor_dim0[31:16] | tensor_dim1=1
    g1[3] = (int)((ndw & 0xFFFFu) << 16);                          // tensor_dim1 hi=0 | tile_dim0=ndw
    g1[4] = 1;                                                     // tile_dim1=1, tile_dim2=0
    g1[5] = (int)ndw;                                              // tensor_dim0_stride[31:0]
    g1[6] = 0;                                                     // stride0 hi | stride1 lo
    g1[7] = 0;                                                     // stride1 hi
    i32x4 gz4; gz4[0] = 0; gz4[1] = 0; gz4[2] = 0; gz4[3] = 0;
    i32x8 gz8;
    gz8[0] = 0; gz8[1] = 0; gz8[2] = 0; gz8[3] = 0;
    gz8[4] = 0; gz8[5] = 0; gz8[6] = 0; gz8[7] = 0;
    __builtin_amdgcn_tensor_load_to_lds(g0, g1, gz4, gz4, gz8, 0);
}
#endif

__device__ __forceinline__ void stage_mesh(const float* verts, const int* faces, const float* uvs,
                                           float* sv, int* sf, float* su,
                                           int ndw_v, int ndw_f, int ndw_u) {
#ifdef HAVE_TDM
    if (threadIdx.x == 0) {  // one wave issues the DMA; TDM ignores EXEC, EXECZ waves skip
        tdm_load_dwords_1d(lds_addr_of(sv), verts, (unsigned)ndw_v);
        tdm_load_dwords_1d(lds_addr_of(sf), faces, (unsigned)ndw_f);
        tdm_load_dwords_1d(lds_addr_of(su), uvs,   (unsigned)ndw_u);
        __builtin_amdgcn_s_wait_tensorcnt(0);
    }
    // Runtime-false write so the compiler can't treat TDM-filled LDS as undef.
    if (ndw_v == -1) { sv[0] = 0.0f; sf[0] = 0; su[0] = 0.0f; }
#else
    for (int i = threadIdx.x; i < ndw_v; i += blockDim.x) sv[i] = verts[i];
    for (int i = threadIdx.x; i < ndw_f; i += blockDim.x) sf[i] = faces[i];
    for (int i = threadIdx.x; i < ndw_u; i += blockDim.x) su[i] = uvs[i];
#endif
    __syncthreads();
}

__device__ __forceinline__ void finalize_pt(F3 p, int bf, float wacc,
                                            const int* sf, const float* sv, const float* su,
                                            float* __restrict__ out, int i) {
    int ia = sf[3 * bf + 0], ib = sf[3 * bf + 1], ic = sf[3 * bf + 2];
    F3 a = f3(sv[3 * ia + 0], sv[3 * ia + 1], sv[3 * ia + 2]);
    F3 b = f3(sv[3 * ib + 0], sv[3 * ib + 1], sv[3 * ib + 2]);
    F3 c = f3(sv[3 * ic + 0], sv[3 * ic + 1], sv[3 * ic + 2]);
    F3 cp = closest_pt(p, a, b, c);
    // Barycentric of cp (raw v,w feed u before clipping, matching reference).
    F3 v0 = sub3(b, a), v1 = sub3(c, a), v2 = sub3(cp, a);
    float d00 = dot3(v0, v0), d01 = dot3(v0, v1), d11 = dot3(v1, v1);
    float d20 = dot3(v2, v0), d21 = dot3(v2, v1);
    float den = safef(d00 * d11 - d01 * d01);
    float rv = (d11 * d20 - d01 * d21) / den;
    float rw = (d00 * d21 - d01 * d20) / den;
    float ru = 1.0f - rv - rw;
    float bu = clamp01(ru), bv = clamp01(rv), bw = clamp01(rw);
    float ux = bu * su[2 * ia + 0] + bv * su[2 * ib + 0] + bw * su[2 * ic + 0];
    float uy = bu * su[2 * ia + 1] + bv * su[2 * ib + 1] + bw * su[2 * ic + 1];
    F3 d = sub3(cp, p);
    float dist = sqrtf(dot3(d, d) + EPSF);
    float winding = wacc * 0.15915494309189535f;  // 2*sum/(4*pi)
    float sd = (winding > 0.5f) ? -dist : dist;
    out[3 * i + 0] = ux;
    out[3 * i + 1] = uy;
    out[3 * i + 2] = sd;
}

__global__ void __launch_bounds__(THREADS)
warpuv_kernel(const float* __restrict__ pts,
              const float* __restrict__ verts,
              const int* __restrict__ faces,
              const float* __restrict__ uvs,
              float* __restrict__ out,
              int npts, int nf, int nv) {
    __shared__ float s_verts[MAX_NV * 3];
    __shared__ int   s_faces[MAX_NF * 3];
    __shared__ float s_uvs[MAX_NV * 2];

    stage_mesh(verts, faces, uvs, s_verts, s_faces, s_uvs, nv * 3, nf * 3, nv * 2);

    int i0 = blockIdx.x * (THREADS * PPT) + (int)threadIdx.x;
    int i1 = i0 + THREADS;
    bool ok0 = i0 < npts, ok1 = i1 < npts;
    F3 p0 = ok0 ? f3(pts[3 * i0], pts[3 * i0 + 1], pts[3 * i0 + 2]) : f3(0.f, 0.f, 0.f);
    F3 p1 = ok1 ? f3(pts[3 * i1], pts[3 * i1 + 1], pts[3 * i1 + 2]) : f3(0.f, 0.f, 0.f);

    float best0 = 3.402823e38f, best1 = 3.402823e38f;
    int bf0 = 0, bf1 = 0;
    float wa0 = 0.0f, wa1 = 0.0f;

    for (int f = 0; f < nf; ++f) {
        // All lanes read the same LDS addresses -> conflict-free broadcast.
        int ia = s_faces[3 * f + 0], ib = s_faces[3 * f + 1], ic = s_faces[3 * f + 2];
        F3 a = f3(s_verts[3 * ia + 0], s_verts[3 * ia + 1], s_verts[3 * ia + 2]);
        F3 b = f3(s_verts[3 * ib + 0], s_verts[3 * ib + 1], s_verts[3 * ib + 2]);
        F3 c = f3(s_verts[3 * ic + 0], s_verts[3 * ic + 1], s_verts[3 * ic + 2]);

        F3 q0 = closest_pt(p0, a, b, c);
        F3 e0 = sub3(q0, p0);
        float dd0 = dot3(e0, e0);
        if (dd0 < best0) { best0 = dd0; bf0 = f; }   // strict < keeps first argmin
        wa0 += wind_term(p0, a, b, c);

        F3 q1 = closest_pt(p1, a, b, c);
        F3 e1 = sub3(q1, p1);
        float dd1 = dot3(e1, e1);
        if (dd1 < best1) { best1 = dd1; bf1 = f; }
        wa1 += wind_term(p1, a, b, c);
    }

    if (ok0) finalize_pt(p0, bf0, wa0, s_faces, s_verts, s_uvs, out, i0);
    if (ok1) finalize_pt(p1, bf1, wa1, s_faces, s_verts, s_uvs, out, i1);
}

// Pure-bandwidth passthrough of rays_directions_world.
__global__ void copy_dirs_kernel(const float4* __restrict__ src, float4* __restrict__ dst, int n4) {
    int i = blockIdx.x * blockDim.x + (int)threadIdx.x;
    if (i < n4) dst[i] = src[i];
}

extern "C" void kernel_launch(void* const* d_in, const int* in_sizes, int n_in,
                              void* d_out, int out_size, void* d_ws, size_t ws_size,
                              hipStream_t stream) {
    const float* pts   = (const float*)d_in[0];   // rays_points_world      (N*3 f32)
    const float* dirs  = (const float*)d_in[1];   // rays_directions_world  (N*3 f32)
    const float* verts = (const float*)d_in[2];   // vertices_posed         (778*3 f32)
    const float* uvs   = (const float*)d_in[3];   // vertex_uvs             (778*2 f32)
    const int*   faces = (const int*)d_in[4];     // faces                  (1538*3 i32)
    float* out = (float*)d_out;

    int npts = in_sizes[0] / 3;
    int nv   = in_sizes[2] / 3;
    int nf   = in_sizes[4] / 3;

    int blocks = (npts + THREADS * PPT - 1) / (THREADS * PPT);
    warpuv_kernel<<<blocks, THREADS, 0, stream>>>(pts, verts, faces, uvs, out, npts, nf, nv);

    // Output layout: [warped (npts*3)] ++ [directions (in_sizes[1])].
    int ndir = in_sizes[1];
    int n4 = ndir / 4;  // 393216 % 4 == 0
    copy_dirs_kernel<<<(n4 + 255) / 256, 256, 0, stream>>>(
        (const float4*)dirs, (float4*)(out + in_sizes[0]), n4);
}